// HyperbolicAttention_65584150610621
// MI455X (gfx1250) — compile-verified
//
#include <hip/hip_runtime.h>
#include <math.h>

// ---------------------------------------------------------------------------
// Hyperbolic graph attention (UHG) for MI455X / gfx1250, wave32.
//  - QKV + output projections: V_WMMA_F32_16X16X4_F32 (exact f32 matrix math)
//  - A-tiles staged in LDS via the Tensor Data Mover (tensor_load_to_lds with
//    a hand-built D# descriptor, padded LDS rows for bank-conflict-free
//    ds_load_b64 fragment reads), s_wait_tensorcnt + barrier.
//  - Edge phase: wave-per-edge gathers, global softmax via encoded atomicMax +
//    exp-sum, segment-sum via HW f32 atomics.
// Workspace requirement: ~213 MB (4 x N*128 f32 + 2E f32 + scalars).
// ---------------------------------------------------------------------------

#define NNODES 100000
#define NEDGES 1000000
#define EPSF   1e-9f
#define LDSPITCH 132   // 128 + 4 DWORD pad per row -> conflict-free ds_load_b64

typedef float v2f __attribute__((ext_vector_type(2)));
typedef float v8f __attribute__((ext_vector_type(8)));
typedef unsigned int u32x4 __attribute__((ext_vector_type(4)));
typedef int i32x4v __attribute__((ext_vector_type(4)));
typedef int i32x8v __attribute__((ext_vector_type(8)));

// ws layout in floats
#define OFF_XP   0UL                 // N*128, later reused as segment-sum accum
#define OFF_Q    12800000UL
#define OFF_K    25600000UL
#define OFF_V    38400000UL
#define OFF_SC   51200000UL          // scores, 2*E
#define OFF_SCAL 53200000UL          // [0]=cr_init [1]=scale [2..3]=maxEnc(u32) [4..5]=sums

#if defined(__HIP_DEVICE_COMPILE__) && __has_builtin(__builtin_amdgcn_tensor_load_to_lds) && __has_builtin(__builtin_amdgcn_s_wait_tensorcnt)
#define HB_TDM 1
#else
#define HB_TDM 0
#endif

// ---------------- helpers ----------------

__device__ __forceinline__ float waveSum32(float v) {
  #pragma unroll
  for (int m = 16; m >= 1; m >>= 1) v += __shfl_xor(v, m, 32);
  return v;
}
// sum within each 16-lane half of the wave
__device__ __forceinline__ float halfSum16(float v) {
  #pragma unroll
  for (int m = 8; m >= 1; m >>= 1) v += __shfl_xor(v, m, 32);
  return v;
}

__device__ __forceinline__ unsigned encf(float f) {
  unsigned u = __float_as_uint(f);
  return (u & 0x80000000u) ? ~u : (u | 0x80000000u);
}
__device__ __forceinline__ float decf(unsigned u) {
  return (u & 0x80000000u) ? __uint_as_float(u & 0x7FFFFFFFu) : __uint_as_float(~u);
}

__device__ __forceinline__ float uhgN(const float* a, const float* b, int n) {
  float s = 0.f;
  for (int i = 0; i < n - 1; ++i) s += a[i] * b[i];
  return s - a[n - 1] * b[n - 1];
}

// Stage `rows` x 128 f32 tile (contiguous rows, row-major) from global into
// LDS with row pitch LDSPITCH, using the Tensor Data Mover when available.
// Must be followed by __syncthreads() by the caller (all threads).
__device__ __forceinline__ void stage_tile(const float* __restrict__ gsrc,
                                           float* __restrict__ lds, int rows) {
#if HB_TDM
  if ((threadIdx.x >> 5) == 0) {   // one wave issues the DMA
    unsigned long long ga = (unsigned long long)(uintptr_t)gsrc;
    unsigned ldsOff = (unsigned)(uintptr_t)lds;
    u32x4 g0;
    g0[0] = 1u;                                   // count=1, is_restore=0
    g0[1] = ldsOff;                               // lds_addr
    g0[2] = (unsigned)(ga & 0xffffffffull);       // global_addr[31:0]
    g0[3] = (unsigned)((ga >> 32) & 0x01ffffffull) | 0x80000000u; // addr[56:32] | type=2
    i32x8v g1;
    // data_size=4B(2), pad_enable=1, pad_interval=6(128 DW), pad_amount=3(4 DW)
    g1[0] = (int)((2u << 16) | (1u << 20) | (6u << 22) | (3u << 25));
    g1[1] = (int)(128u << 16);                    // tensor_dim0 = 128 (lo16<<16)
    g1[2] = (int)((unsigned)rows << 16);          // tensor_dim1 = rows (lo16<<16)
    g1[3] = (int)(128u << 16);                    // tile_dim0 = 128
    g1[4] = rows;                                 // tile_dim1 = rows, tile_dim2 = 0
    g1[5] = 128;                                  // tensor_dim0_stride = 128
    g1[6] = 0;
    g1[7] = 0;
    i32x4v gz = {0, 0, 0, 0};
#if defined(__clang_major__) && (__clang_major__ >= 23)
    i32x8v gz8 = {0, 0, 0, 0, 0, 0, 0, 0};
    __builtin_amdgcn_tensor_load_to_lds(g0, g1, gz, gz, gz8, 0);
#else
    __builtin_amdgcn_tensor_load_to_lds(g0, g1, gz, gz, 0);
#endif
    __builtin_amdgcn_s_wait_tensorcnt(0);
  }
#else
  // cooperative copy with manual row padding
  const float4* src = (const float4*)gsrc;
  const int nf4 = rows * 32;                      // 128 floats = 32 float4 per row
  for (int i = threadIdx.x; i < nf4; i += blockDim.x) {
    const int r = i >> 5, c4 = i & 31;
    ((float4*)(lds + r * LDSPITCH))[c4] = src[i];
  }
#endif
}

// ---------------- kernels ----------------

// cr_initial from raw x rows 0..3; init softmax scalars.
__global__ void hb_init(const float* __restrict__ x, float* __restrict__ scal) {
  if (threadIdx.x == 0 && blockIdx.x == 0) {
    const float *p0 = x, *p1 = x + 128, *p2 = x + 256, *p3 = x + 384;
    float num = uhgN(p0, p2, 128) * uhgN(p1, p3, 128);
    float den = uhgN(p0, p3, 128) * uhgN(p1, p2, 128);
    den = (fabsf(den) < EPSF) ? EPSF : den;
    scal[0] = num / den;
    ((unsigned*)scal)[2] = 0x00800000u;  // enc(-FLT_MAX)
    ((unsigned*)scal)[3] = 0x00800000u;
    scal[4] = 0.f; scal[5] = 0.f;
  }
}

// projective normalize of x: unit-norm first 127 dims, keep dim 127. wave/node.
__global__ void hb_norm_x(const float* __restrict__ x, float* __restrict__ xp) {
  const int wave = threadIdx.x >> 5, lane = threadIdx.x & 31;
  const int n = blockIdx.x * 8 + wave;
  float4 vv = reinterpret_cast<const float4*>(x + (size_t)n * 128)[lane];
  float s = vv.x * vv.x + vv.y * vv.y + vv.z * vv.z;
  if (lane != 31) s += vv.w * vv.w;              // exclude homogeneous coord 127
  s = waveSum32(s);
  float inv = 1.0f / fmaxf(sqrtf(s), EPSF);
  vv.x *= inv; vv.y *= inv; vv.z *= inv;
  if (lane != 31) vv.w *= inv;
  reinterpret_cast<float4*>(xp + (size_t)n * 128)[lane] = vv;
}

// fused QKV GEMM: [N,128] @ W^T (+bias) with f32 WMMA.
// A tile (16x128) staged to LDS once per block (TDM), reused by all 8 waves;
// software-pipelined K loop: next frags load while current WMMAs execute.
__global__ void hb_gemm_qkv(const float* __restrict__ xp,
                            const float* __restrict__ Wq, const float* __restrict__ bq,
                            const float* __restrict__ Wk, const float* __restrict__ bk,
                            const float* __restrict__ Wv, const float* __restrict__ bv,
                            float* __restrict__ q, float* __restrict__ k,
                            float* __restrict__ v) {
  __shared__ float tileA[16 * LDSPITCH];
  const int wave = threadIdx.x >> 5, lane = threadIdx.x & 31;
  const int lr = lane & 15, half = lane >> 4;
  const int m0 = blockIdx.x * 16;
  const int h0 = wave * 16;

  stage_tile(xp + (size_t)m0 * 128, tileA, 16);
  __syncthreads();

  const float* As = tileA + lr * LDSPITCH + 2 * half;
  const float* Bq = Wq + (size_t)(h0 + lr) * 128 + 2 * half;
  const float* Bk = Wk + (size_t)(h0 + lr) * 128 + 2 * half;
  const float* Bv = Wv + (size_t)(h0 + lr) * 128 + 2 * half;

  v8f c0 = {}, c1 = {}, c2 = {};
  float2 a  = *reinterpret_cast<const float2*>(As);
  float2 b0 = *reinterpret_cast<const float2*>(Bq);
  float2 b1 = *reinterpret_cast<const float2*>(Bk);
  float2 b2 = *reinterpret_cast<const float2*>(Bv);
  #pragma unroll
  for (int kk = 0; kk < 128; kk += 4) {
    float2 an = a, b0n = b0, b1n = b1, b2n = b2;
    if (kk + 4 < 128) {                         // prefetch next step's frags
      an  = *reinterpret_cast<const float2*>(As + kk + 4);
      b0n = *reinterpret_cast<const float2*>(Bq + kk + 4);
      b1n = *reinterpret_cast<const float2*>(Bk + kk + 4);
      b2n = *reinterpret_cast<const float2*>(Bv + kk + 4);
    }
    v2f av = {a.x, a.y};
    v2f w0 = {b0.x, b0.y}, w1 = {b1.x, b1.y}, w2 = {b2.x, b2.y};
    c0 = __builtin_amdgcn_wmma_f32_16x16x4_f32(false, av, false, w0, (short)0, c0, false, false);
    c1 = __builtin_amdgcn_wmma_f32_16x16x4_f32(false, av, false, w1, (short)0, c1, false, false);
    c2 = __builtin_amdgcn_wmma_f32_16x16x4_f32(false, av, false, w2, (short)0, c2, false, false);
    a = an; b0 = b0n; b1 = b1n; b2 = b2n;
  }
  #pragma unroll
  for (int j = 0; j < 8; ++j) {
    const int row = m0 + j + 8 * half;
    const int col = h0 + lr;
    q[(size_t)row * 128 + col] = c0[j] + bq[col];
    k[(size_t)row * 128 + col] = c1[j] + bk[col];
    v[(size_t)row * 128 + col] = c2[j] + bv[col];
  }
}

__global__ void hb_zero(float* __restrict__ p, long n) {
  long i = blockIdx.x * (long)blockDim.x + threadIdx.x;
  if (i < n) p[i] = 0.f;
}

// per-head projective normalize of q and k in place. wave/node; head0=lanes 0-15.
__global__ void hb_norm_qk(float* __restrict__ q, float* __restrict__ k) {
  const int wave = threadIdx.x >> 5, lane = threadIdx.x & 31;
  const int n = blockIdx.x * 8 + wave;
  const bool timeLane = (lane == 15) || (lane == 31);  // holds coord 63 / 127 in .w
  #pragma unroll
  for (int which = 0; which < 2; ++which) {
    float* p = which ? k : q;
    float4 vv = reinterpret_cast<float4*>(p + (size_t)n * 128)[lane];
    float s = vv.x * vv.x + vv.y * vv.y + vv.z * vv.z + (timeLane ? 0.f : vv.w * vv.w);
    s = halfSum16(s);
    float inv = 1.0f / fmaxf(sqrtf(s), EPSF);
    vv.x *= inv; vv.y *= inv; vv.z *= inv;
    if (!timeLane) vv.w *= inv;
    reinterpret_cast<float4*>(p + (size_t)n * 128)[lane] = vv;
  }
}

// per-edge hyperbolic scores + global max (encoded uint atomicMax). wave/edge.
__global__ void hb_scores(const float* __restrict__ qn, const float* __restrict__ kn,
                          const int* __restrict__ ei, float* __restrict__ scores,
                          unsigned* __restrict__ maxEnc) {
  const int wave = threadIdx.x >> 5, lane = threadIdx.x & 31;
  const int e = blockIdx.x * 8 + wave;
  const int row = ei[e], col = ei[NEDGES + e];
  float4 qv = reinterpret_cast<const float4*>(qn + (size_t)row * 128)[lane];
  float4 kv = reinterpret_cast<const float4*>(kn + (size_t)col * 128)[lane];
  const bool timeLane = (lane == 15) || (lane == 31);
  float s = qv.x * kv.x + qv.y * kv.y + qv.z * kv.z;
  s += (timeLane ? -1.f : 1.f) * qv.w * kv.w;
  s = halfSum16(s);
  s *= 0.08838834764831845f;  // 1/sqrt(128)
  __shared__ float smax[2][8];
  if (lane == 0)  { scores[2 * e + 0] = s; smax[0][wave] = s; }
  if (lane == 16) { scores[2 * e + 1] = s; smax[1][wave] = s; }
  __syncthreads();
  if (threadIdx.x < 2) {
    float m = smax[threadIdx.x][0];
    #pragma unroll
    for (int i = 1; i < 8; ++i) m = fmaxf(m, smax[threadIdx.x][i]);
    atomicMax(&maxEnc[threadIdx.x], encf(m));
  }
}

// sum of exp(score - max) per head over all 2E scores (streaming, nontemporal).
__global__ void hb_expsum(const float* __restrict__ scores,
                          const unsigned* __restrict__ maxEnc,
                          float* __restrict__ sums) {
  const long i = blockIdx.x * (long)blockDim.x + threadIdx.x;
  const int lane = threadIdx.x & 31, wave = threadIdx.x >> 5;
  const float m0 = decf(maxEnc[0]), m1 = decf(maxEnc[1]);
  float val = 0.f;
  if (i < 2L * NEDGES) {
    const int h = (int)(i & 1);
    val = expf(__builtin_nontemporal_load(&scores[i]) - (h ? m1 : m0));
  }
  // parity-preserving wave reduction: lane0 <- even lanes (h0), lane1 <- odd (h1)
  #pragma unroll
  for (int m = 16; m >= 2; m >>= 1) val += __shfl_xor(val, m, 32);
  __shared__ float ssum[2][8];
  if (lane < 2) ssum[lane][wave] = val;
  __syncthreads();
  if (threadIdx.x < 2) {
    float t = ssum[threadIdx.x][0];
    #pragma unroll
    for (int w = 1; w < 8; ++w) t += ssum[threadIdx.x][w];
    atomicAdd(&sums[threadIdx.x], t);
  }
}

// messages + segment_sum via f32 HW atomics. wave/edge, float4/lane.
__global__ void hb_msgs(const float* __restrict__ vbuf, const float* __restrict__ scores,
                        const int* __restrict__ ei, const unsigned* __restrict__ maxEnc,
                        const float* __restrict__ sums, float* __restrict__ accum) {
  const int wave = threadIdx.x >> 5, lane = threadIdx.x & 31;
  const int e = blockIdx.x * 8 + wave;
  const int row = ei[e], col = ei[NEDGES + e];
  const int h = lane >> 4;
  const float w = expf(scores[2 * e + h] - decf(maxEnc[h])) / sums[h];
  float4 vv = reinterpret_cast<const float4*>(vbuf + (size_t)col * 128)[lane];
  float* dst = accum + (size_t)row * 128 + 4 * lane;
  unsafeAtomicAdd(dst + 0, vv.x * w);
  unsafeAtomicAdd(dst + 1, vv.y * w);
  unsafeAtomicAdd(dst + 2, vv.z * w);
  unsafeAtomicAdd(dst + 3, vv.w * w);
}

// output GEMM: [N,128] @ Wo^T (+bo) -> [N,64], f32 WMMA, 32-row LDS tile.
__global__ void hb_gemm_out(const float* __restrict__ accum, const float* __restrict__ Wo,
                            const float* __restrict__ bo, float* __restrict__ out) {
  __shared__ float tileA[32 * LDSPITCH];
  const int wave = threadIdx.x >> 5, lane = threadIdx.x & 31;
  const int lr = lane & 15, half = lane >> 4;
  const int rt = wave >> 2, ct = wave & 3;
  const int m0 = blockIdx.x * 32 + rt * 16;
  const int h0 = ct * 16;

  stage_tile(accum + (size_t)blockIdx.x * 32 * 128, tileA, 32);
  __syncthreads();

  const float* As = tileA + (rt * 16 + lr) * LDSPITCH + 2 * half;
  const float* B  = Wo + (size_t)(h0 + lr) * 128 + 2 * half;

  v8f c = {};
  float2 a = *reinterpret_cast<const float2*>(As);
  float2 b = *reinterpret_cast<const float2*>(B);
  #pragma unroll
  for (int kk = 0; kk < 128; kk += 4) {
    float2 an = a, bn = b;
    if (kk + 4 < 128) {
      an = *reinterpret_cast<const float2*>(As + kk + 4);
      bn = *reinterpret_cast<const float2*>(B + kk + 4);
    }
    v2f av = {a.x, a.y};
    v2f bv = {b.x, b.y};
    c = __builtin_amdgcn_wmma_f32_16x16x4_f32(false, av, false, bv, (short)0, c, false, false);
    a = an; b = bn;
  }
  #pragma unroll
  for (int j = 0; j < 8; ++j) {
    const int row = m0 + j + 8 * half;
    const int col = h0 + lr;
    out[(size_t)row * 64 + col] = c[j] + bo[col];
  }
}

// scale = |cr_target / cr_cur|^0.25 from out rows 0..3 (pre-scale).
__global__ void hb_cr(const float* __restrict__ out, float* __restrict__ scal) {
  if (threadIdx.x == 0 && blockIdx.x == 0) {
    const float *p0 = out, *p1 = out + 64, *p2 = out + 128, *p3 = out + 192;
    float num = uhgN(p0, p2, 64) * uhgN(p1, p3, 64);
    float den = uhgN(p0, p3, 64) * uhgN(p1, p2, 64);
    den = (fabsf(den) < EPSF) ? EPSF : den;
    float cr = num / den;
    cr = (fabsf(cr) < EPSF) ? EPSF : cr;
    scal[1] = powf(fabsf(scal[0] / cr), 0.25f);
  }
}

__global__ void hb_scale(float* __restrict__ out, const float* __restrict__ scal) {
  long i = blockIdx.x * (long)blockDim.x + threadIdx.x;
  if (i < (long)NNODES * 64) out[i] *= scal[1];
}

// ---------------- launch ----------------

extern "C" void kernel_launch(void* const* d_in, const int* in_sizes, int n_in,
                              void* d_out, int out_size, void* d_ws, size_t ws_size,
                              hipStream_t stream) {
  (void)in_sizes; (void)n_in; (void)out_size; (void)ws_size;
  const float* x  = (const float*)d_in[0];
  const int*   ei = (const int*)d_in[1];
  const float* Wq = (const float*)d_in[2]; const float* bq = (const float*)d_in[3];
  const float* Wk = (const float*)d_in[4]; const float* bk = (const float*)d_in[5];
  const float* Wv = (const float*)d_in[6]; const float* bv = (const float*)d_in[7];
  const float* Wo = (const float*)d_in[8]; const float* bo = (const float*)d_in[9];
  float* out = (float*)d_out;
  float* ws  = (float*)d_ws;

  float* xp    = ws + OFF_XP;   // aliased: x_proj, then segment-sum accumulator
  float* q     = ws + OFF_Q;
  float* k     = ws + OFF_K;
  float* v     = ws + OFF_V;
  float* sc    = ws + OFF_SC;
  float* scal  = ws + OFF_SCAL;
  unsigned* maxEnc = (unsigned*)scal + 2;
  float* sums  = scal + 4;

  hb_init    <<<1, 32, 0, stream>>>(x, scal);
  hb_norm_x  <<<NNODES / 8, 256, 0, stream>>>(x, xp);
  hb_gemm_qkv<<<NNODES / 16, 256, 0, stream>>>(xp, Wq, bq, Wk, bk, Wv, bv, q, k, v);
  hb_zero    <<<(NNODES * 128) / 256, 256, 0, stream>>>(xp, (long)NNODES * 128);
  hb_norm_qk <<<NNODES / 8, 256, 0, stream>>>(q, k);
  hb_scores  <<<NEDGES / 8, 256, 0, stream>>>(q, k, ei, sc, maxEnc);
  hb_expsum  <<<(2 * NEDGES + 255) / 256, 256, 0, stream>>>(sc, maxEnc, sums);
  hb_msgs    <<<NEDGES / 8, 256, 0, stream>>>(v, sc, ei, maxEnc, sums, xp);
  hb_gemm_out<<<NNODES / 32, 256, 0, stream>>>(xp, Wo, bo, out);
  hb_cr      <<<1, 32, 0, stream>>>(out, scal);
  hb_scale   <<<(NNODES * 64) / 256, 256, 0, stream>>>(out, scal);
}